// SinkhornMatchingLoss_62457414418466
// MI455X (gfx1250) — compile-verified
//
#include <hip/hip_runtime.h>
#include <hip/hip_bf16.h>

typedef __attribute__((ext_vector_type(2))) float v2f;
typedef __attribute__((ext_vector_type(8))) float v8f;

#define NB 256          // N (points per frame)
#define DD 128          // D (feature dim)
#define TAU_ 0.05f
#define EPS_ 1e-8f
#define N_ITER 10

// LDS layout (in floats)
#define K_OFF    0
#define S2_OFF   (NB * NB)
#define T2_OFF   (S2_OFF + NB)
#define U_OFF    (T2_OFF + NB)
#define V_OFF    (U_OFF + NB)
#define PART_OFF (V_OFF + NB)
#define RED_OFF  (PART_OFF + 4 * NB)
#define LDS_FLOATS (RED_OFF + 32)

__global__ __launch_bounds__(1024) void sinkhorn_bt_kernel(
    const float* __restrict__ src, const float* __restrict__ tgt,
    float* __restrict__ out, float inv_bt)
{
    extern __shared__ float lds[];
    const int bt   = blockIdx.x;
    const float* S = src + (size_t)bt * NB * DD;
    const float* T = tgt + (size_t)bt * NB * DD;
    const int tid  = threadIdx.x;
    const int lane = tid & 31;
    const int wave = tid >> 5;
    const int half = lane >> 4;   // 0 or 1
    const int l16  = lane & 15;

    // ---- per-row squared norms into LDS ----
    if (tid < 2 * NB) {
        const bool is_s = (tid < NB);
        const int  row  = is_s ? tid : (tid - NB);
        const float4* p4 = (const float4*)((is_s ? S : T) + row * DD);
        float acc = 0.0f;
        #pragma unroll 8
        for (int i = 0; i < DD / 4; ++i) {
            float4 q = p4[i];
            acc += q.x * q.x + q.y * q.y + q.z * q.z + q.w * q.w;
        }
        lds[(is_s ? S2_OFF : T2_OFF) + row] = acc;
    }
    __syncthreads();

    // ---- K = exp(-cdist/tau) via V_WMMA_F32_16X16X4_F32, K resident in LDS ----
    // 16x16 grid of 16x16 tiles = 256 tiles, 32 waves -> 8 tiles per wave.
    // Uniform control flow across the block: EXEC all-ones for WMMA.
    for (int t = wave; t < 256; t += 32) {
        const int ti = t >> 4;
        const int tj = t & 15;
        const int arow = ti * 16 + l16;   // source row for A fragment
        const int bcol = tj * 16 + l16;   // target row (B column) for B fragment
        v8f acc = {};
        #pragma unroll 8
        for (int kk = 0; kk < DD / 4; ++kk) {
            const int d = kk * 4 + 2 * half;   // K index = vgpr + 2*half
            v2f a = *(const v2f*)(S + arow * DD + d);
            v2f b = *(const v2f*)(T + bcol * DD + d);
            // D = A(16x4) x B(4x16) + C, fp32 throughout
            acc = __builtin_amdgcn_wmma_f32_16x16x4_f32(
                false, a, false, b, (short)0, acc, false, false);
        }
        const int m = tj * 16 + l16;
        #pragma unroll
        for (int r = 0; r < 8; ++r) {
            const int n = ti * 16 + r + 8 * half;
            float c2   = lds[S2_OFF + n] + lds[T2_OFF + m] - 2.0f * acc[r];
            float cst  = sqrtf(fmaxf(c2, 0.0f));
            lds[K_OFF + n * NB + m] = expf(-cst * (1.0f / TAU_));
        }
    }
    if (tid < NB) { lds[U_OFF + tid] = 1.0f; lds[V_OFF + tid] = 1.0f; }
    __syncthreads();

    // ---- 10 Sinkhorn iterations, K in LDS ----
    const float MU = 1.0f / (float)NB;
    const int rc = tid & (NB - 1);   // row/col handled by this thread
    const int ch = tid >> 8;         // chunk 0..3 (64 elems each)
    for (int it = 0; it < N_ITER; ++it) {
        // u = MU / (K v + eps)
        {
            const float* Krow = lds + K_OFF + rc * NB + ch * 64;
            const float* vv   = lds + V_OFF + ch * 64;
            float s = 0.0f;
            #pragma unroll 8
            for (int j = 0; j < 64; ++j) s += Krow[j] * vv[j];
            lds[PART_OFF + ch * NB + rc] = s;
        }
        __syncthreads();
        if (tid < NB) {
            float s = lds[PART_OFF + tid] + lds[PART_OFF + NB + tid]
                    + lds[PART_OFF + 2 * NB + tid] + lds[PART_OFF + 3 * NB + tid];
            lds[U_OFF + tid] = MU / (s + EPS_);
        }
        __syncthreads();
        // v = MU / (K^T u + eps)  (column walk: lanes hit distinct banks)
        {
            float s = 0.0f;
            #pragma unroll 8
            for (int i = 0; i < 64; ++i)
                s += lds[K_OFF + (ch * 64 + i) * NB + rc] * lds[U_OFF + ch * 64 + i];
            lds[PART_OFF + ch * NB + rc] = s;
        }
        __syncthreads();
        if (tid < NB) {
            float s = lds[PART_OFF + tid] + lds[PART_OFF + NB + tid]
                    + lds[PART_OFF + 2 * NB + tid] + lds[PART_OFF + 3 * NB + tid];
            lds[V_OFF + tid] = MU / (s + EPS_);
        }
        __syncthreads();
    }

    // ---- total = sum u[n] * K[n,m] * v[m] * cost[n,m], cost = -tau*ln(K) ----
    float acc = 0.0f;
    for (int idx = tid; idx < NB * NB; idx += 1024) {
        const int n = idx >> 8;
        const int m = idx & (NB - 1);
        float k = lds[K_OFF + idx];
        if (k > 1e-30f) {
            float cst = -TAU_ * logf(k);
            acc += lds[U_OFF + n] * k * lds[V_OFF + m] * cst;
        }
    }
    #pragma unroll
    for (int off = 16; off > 0; off >>= 1) acc += __shfl_xor(acc, off, 32);
    if (lane == 0) lds[RED_OFF + wave] = acc;
    __syncthreads();
    if (wave == 0) {
        float s = lds[RED_OFF + lane];   // 32 waves -> 32 lanes, exact fit
        #pragma unroll
        for (int off = 16; off > 0; off >>= 1) s += __shfl_xor(s, off, 32);
        if (lane == 0) atomicAdd(out, s * inv_bt);
    }
}

extern "C" void kernel_launch(void* const* d_in, const int* in_sizes, int n_in,
                              void* d_out, int out_size, void* d_ws, size_t ws_size,
                              hipStream_t stream) {
    const float* src = (const float*)d_in[0];
    const float* tgt = (const float*)d_in[1];
    float* out = (float*)d_out;

    const int BT = 16 * 64;  // B*T independent problems
    const size_t lds_bytes = (size_t)LDS_FLOATS * sizeof(float);  // ~264 KB < 320 KB/WGP

    hipMemsetAsync(out, 0, sizeof(float) * (out_size > 0 ? out_size : 1), stream);
    hipFuncSetAttribute((const void*)sinkhorn_bt_kernel,
                        hipFuncAttributeMaxDynamicSharedMemorySize, (int)lds_bytes);
    hipLaunchKernelGGL(sinkhorn_bt_kernel, dim3(BT), dim3(1024), lds_bytes, stream,
                       src, tgt, out, 1.0f / (float)BT);
}